// LatentSDE_41635412968039
// MI455X (gfx1250) — compile-verified
//
#include <hip/hip_runtime.h>
#include <hip/hip_bf16.h>
#include <math.h>

#define T_STEPS 128
#define BATCH   2048
#define HID     256
#define CTXC    128
#define LAT     32
#define INF     (LAT + CTXC)   /* 160 */
#define LOG2PI_F 1.8378770664093453f
#define KMAX    256
#define KPAD    4              /* LDS row padding (floats) for bank spread + 16B align */

typedef float v2f __attribute__((ext_vector_type(2)));
typedef float v4f __attribute__((ext_vector_type(4)));
typedef float v8f __attribute__((ext_vector_type(8)));

__device__ __forceinline__ float softplus_f(float x) {
    return (x > 20.0f) ? x : log1pf(__expf(x));
}
__device__ __forceinline__ float sigmoid_f(float x) {
    return 1.0f / (1.0f + __expf(-x));
}
__device__ __forceinline__ unsigned hash_u32(unsigned x) {
    x ^= x >> 17; x *= 0xed5ad4bbu;
    x ^= x >> 11; x *= 0xac4c1b51u;
    x ^= x >> 15; x *= 0x31848babu;
    x ^= x >> 14;
    return x;
}
__device__ __forceinline__ float rand_normal(unsigned idx, unsigned salt) {
    unsigned a = hash_u32(idx * 2654435761u + salt);
    unsigned b = hash_u32(a ^ 0x9e3779b9u);
    float u1 = ((a >> 8) + 1u) * (1.0f / 16777217.0f);   // (0,1]
    float u2 = (b >> 8) * (1.0f / 16777216.0f);          // [0,1)
    return sqrtf(-2.0f * __logf(u1)) * __cosf(6.2831853071795864f * u2);
}

__device__ __forceinline__ v8f wmma4(v2f a, v2f b, v8f c) {
    // V_WMMA_F32_16X16X4_F32: D = A(16x4,f32) x B(4x16,f32) + C(16x16,f32)
    return __builtin_amdgcn_wmma_f32_16x16x4_f32(
        false, a, false, b, (short)0, c, false, false);
}

// ---------------------------------------------------------------------------
// C[M, NT*16*gridX] = act( A[M,K] @ W[N,K]^T + bias[N] )
// Block = 128 threads = 4 waves. Block tile: 64 rows x (NT*16) cols.
// Both the A panel (64 x K) and W panel (NT*16 x K) are staged in LDS once
// (coalesced b128 global loads), then the K loop runs branch-free out of LDS.
// Requires: M % 64 == 0, K % 4 == 0, bias != nullptr. EXEC all-ones always.
// ---------------------------------------------------------------------------
template <int NT, bool SOFTPLUS>
__global__ __launch_bounds__(128)
void gemm_tile(const float* __restrict__ A, int lda,
               const float* __restrict__ W, int ldw,
               const float* __restrict__ bias,
               float* __restrict__ C, int ldc, int K)
{
    __shared__ float smem[(64 + 64) * (KMAX + KPAD)];   // worst case 133 KB (<320 KB/WGP)

    const int tid  = threadIdx.x;
    const int lane = tid & 31;
    const int wave = tid >> 5;
    const int half = lane >> 4;       // lanes 0-15 vs 16-31
    const int lr   = lane & 15;
    const int kp   = K + KPAD;
    const int n00  = blockIdx.x * (NT * 16);
    const int mBlk = blockIdx.y * 64;

    float* sW = smem;                     // [NT*16][kp]
    float* sA = smem + NT * 16 * kp;      // [64][kp]

    // ---- cooperative staging (each element read once per block) ----
    const int kq = K >> 2;                // float4s per row
    for (int i = tid; i < NT * 16 * kq; i += 128) {
        int r = i / kq, c = (i - r * kq) * 4;
        *(v4f*)(sW + r * kp + c) = *(const v4f*)(W + (size_t)(n00 + r) * ldw + c);
    }
    for (int i = tid; i < 64 * kq; i += 128) {
        int r = i / kq, c = (i - r * kq) * 4;
        *(v4f*)(sA + r * kp + c) = *(const v4f*)(A + (size_t)(mBlk + r) * lda + c);
    }
    __syncthreads();

    // ---- branch-free WMMA K loop from LDS ----
    // A frag: lane(half,lr) holds A[m0+lr, k+2*half +{0,1}]
    // B frag: lane(half,lr) holds W[n0+lr, k+2*half +{0,1}]
    const float* a_row  = sA + (wave * 16 + lr) * kp + 2 * half;
    const float* w_base = sW + lr * kp + 2 * half;

    v8f acc[NT];
    #pragma unroll
    for (int t = 0; t < NT; ++t) acc[t] = (v8f){0.f,0.f,0.f,0.f,0.f,0.f,0.f,0.f};

    #pragma unroll 4
    for (int k = 0; k < K; k += 4) {
        v2f a = *(const v2f*)(a_row + k);
        #pragma unroll
        for (int t = 0; t < NT; ++t) {
            v2f b = *(const v2f*)(w_base + t * 16 * kp + k);
            acc[t] = wmma4(a, b, acc[t]);
        }
    }

    // ---- epilogue: C/D layout VGPR v -> row m0 + v + 8*half, col n0 + lr ----
    #pragma unroll
    for (int t = 0; t < NT; ++t) {
        const int n0 = n00 + t * 16;
        const float bv = bias[n0 + lr];
        #pragma unroll
        for (int v = 0; v < 8; ++v) {
            float x = acc[t][v] + bv;
            if (SOFTPLUS) x = softplus_f(x);
            C[(size_t)(mBlk + wave * 16 + v + 8 * half) * ldc + n0 + lr] = x;
        }
    }
}

// ---------------------------------------------------------------------------
__global__ void zero_buf(float* __restrict__ p, int n) {
    int i = blockIdx.x * blockDim.x + threadIdx.x;
    if (i < n) p[i] = 0.0f;
}

// GRU gate fuse: xp computed on-the-fly (D == 1), gh already holds h@Whh.T+bhh
__global__ void gru_gate(const float* __restrict__ xs, int t,
                         const float* __restrict__ gh,
                         const float* __restrict__ hprev,
                         const float* __restrict__ Wih,
                         const float* __restrict__ bih,
                         float* __restrict__ hnew)
{
    int idx = blockIdx.x * blockDim.x + threadIdx.x;
    if (idx >= BATCH * HID) return;
    int b = idx / HID, j = idx % HID;
    float x = xs[(size_t)(T_STEPS - 1 - t) * BATCH + b];   // reversed input
    const float* ghb = gh + (size_t)b * 3 * HID;
    float r  = sigmoid_f(x * Wih[j]         + bih[j]         + ghb[j]);
    float zg = sigmoid_f(x * Wih[HID + j]   + bih[HID + j]   + ghb[HID + j]);
    float n  = tanhf(    x * Wih[2*HID + j] + bih[2*HID + j] + r * ghb[2*HID + j]);
    hnew[idx] = (1.0f - zg) * n + zg * hprev[idx];
}

__global__ void concat_zc(const float* __restrict__ z, const float* __restrict__ ctx_t,
                          float* __restrict__ cat)
{
    int idx = blockIdx.x * blockDim.x + threadIdx.x;
    if (idx >= BATCH * INF) return;
    int b = idx / INF, j = idx % INF;
    cat[idx] = (j < LAT) ? z[(size_t)b * LAT + j] : ctx_t[(size_t)b * CTXC + (j - LAT)];
}

// z0 = mean + exp(logstd)*eps ; also init accumulators and logp(t=0)
__global__ void init_z0(const float* __restrict__ q, float* __restrict__ z,
                        float* __restrict__ alr, float* __restrict__ apen,
                        float* __restrict__ alp,
                        const float* __restrict__ xs, const float* __restrict__ noise)
{
    int idx = blockIdx.x * blockDim.x + threadIdx.x;
    if (idx >= BATCH * LAT) return;
    int b = idx / LAT, l = idx % LAT;
    float mean = q[(size_t)b * 2 * LAT + l];
    float lstd = q[(size_t)b * 2 * LAT + LAT + l];
    float eps  = rand_normal((unsigned)idx, 0xA5E5u);
    float zv   = mean + __expf(lstd) * eps;
    z[idx] = zv;
    if (l == 0) {
        alr[b] = 0.0f; apen[b] = 0.0f;
        float s = noise[0];
        float d = (xs[b] - zv) / s;
        alp[b] = -0.5f * d * d - __logf(s) - 0.5f * LOG2PI_F;
    }
}

// g_diag: thread layout l*B+b so a wave shares l (broadcast-friendly weight reads)
__global__ void gdiag_kernel(const float* __restrict__ z,
                             const float* __restrict__ gW1, const float* __restrict__ gb1,
                             const float* __restrict__ gW2, const float* __restrict__ gb2,
                             const float* __restrict__ gW3, const float* __restrict__ gb3,
                             float* __restrict__ g)
{
    int idx = blockIdx.x * blockDim.x + threadIdx.x;
    if (idx >= BATCH * LAT) return;
    int l = idx / BATCH, b = idx % BATCH;
    float zv = z[(size_t)b * LAT + l];
    const float* w1 = gW1 + (size_t)l * HID;
    const float* c1 = gb1 + (size_t)l * HID;
    const float* w2 = gW2 + (size_t)l * HID;
    float acc = 0.0f;
    #pragma unroll 4
    for (int h = 0; h < HID; ++h)
        acc += softplus_f(zv * w1[h] + c1[h]) * w2[h];
    float s = sigmoid_f(acc + gb2[l]);
    g[(size_t)b * LAT + l] = softplus_f(s * gW3[l] + gb3[l]);
}

// Euler-Maruyama step + KL-path / noise / likelihood accumulation
__global__ void sde_combine(const float* __restrict__ f, const float* __restrict__ hdr,
                            const float* __restrict__ g,
                            const float* __restrict__ zcur, float* __restrict__ znext,
                            const float* __restrict__ ts, int t,
                            const float* __restrict__ xs, const float* __restrict__ noise,
                            float* __restrict__ alr, float* __restrict__ apen,
                            float* __restrict__ alp)
{
    int b = blockIdx.x * blockDim.x + threadIdx.x;
    if (b >= BATCH) return;
    float dt  = ts[t + 1] - ts[t];
    float sdt = sqrtf(dt);
    float su = 0.0f, sg = 0.0f, zn0 = 0.0f;
    #pragma unroll 4
    for (int l = 0; l < LAT; ++l) {
        size_t i = (size_t)b * LAT + l;
        float fv = f[i], hv = hdr[i], gv = g[i], zv = zcur[i];
        float u = (fv - hv) / gv;
        su += u * u;
        sg += gv * gv;
        float dw = rand_normal((unsigned)(((unsigned)t * BATCH + (unsigned)b) * LAT + (unsigned)l), 0xD17Au);
        float zn = zv + fv * dt + gv * sdt * dw;
        znext[i] = zn;
        if (l == 0) zn0 = zn;
    }
    alr[b]  += 0.5f * su * dt;
    apen[b] += sg * dt;
    float s = noise[0];
    float d = (xs[(size_t)(t + 1) * BATCH + b] - zn0) / s;
    alp[b]  += -0.5f * d * d - __logf(s) - 0.5f * LOG2PI_F;
}

__global__ void final_reduce(const float* __restrict__ alp, const float* __restrict__ alr,
                             const float* __restrict__ apen, const float* __restrict__ q,
                             const float* __restrict__ pz0m, const float* __restrict__ pz0l,
                             const float* __restrict__ ts, float* __restrict__ out)
{
    __shared__ float s1[256], s2[256], s3[256];
    int tid = threadIdx.x;
    float a = 0.0f, c = 0.0f, d = 0.0f, kl = 0.0f;
    for (int b = tid; b < BATCH; b += 256) {
        a += alp[b]; c += alr[b]; d += apen[b];
        for (int l = 0; l < LAT; ++l) {
            float qm = q[(size_t)b * 2 * LAT + l];
            float ql = q[(size_t)b * 2 * LAT + LAT + l];
            float pm = pz0m[l], pl = pz0l[l];
            float dm = qm - pm;
            kl += pl - ql + (__expf(2.0f * ql) + dm * dm) / (2.0f * __expf(2.0f * pl)) - 0.5f;
        }
    }
    s1[tid] = a; s2[tid] = c + kl; s3[tid] = d;
    __syncthreads();
    for (int st = 128; st > 0; st >>= 1) {
        if (tid < st) { s1[tid] += s1[tid+st]; s2[tid] += s2[tid+st]; s3[tid] += s3[tid+st]; }
        __syncthreads();
    }
    if (tid == 0) {
        float dt0 = ts[1] - ts[0];
        out[0] = dt0 * (s1[0] / (float)BATCH);   // log_pxs
        out[1] = s2[0] / (float)BATCH;           // logqp0 + logqp_path
        out[2] = s3[0] / (float)BATCH;           // noise penalty
    }
}

// ---------------------------------------------------------------------------
extern "C" void kernel_launch(void* const* d_in, const int* in_sizes, int n_in,
                              void* d_out, int out_size, void* d_ws, size_t ws_size,
                              hipStream_t stream)
{
    (void)in_sizes; (void)n_in; (void)out_size; (void)ws_size;
    const float* xs    = (const float*)d_in[0];
    const float* ts    = (const float*)d_in[1];
    const float* noise = (const float*)d_in[2];
    const float* Wih   = (const float*)d_in[3];
    const float* Whh   = (const float*)d_in[4];
    const float* bih   = (const float*)d_in[5];
    const float* bhh   = (const float*)d_in[6];
    const float* encW  = (const float*)d_in[7];
    const float* encB  = (const float*)d_in[8];
    const float* qW    = (const float*)d_in[9];
    const float* qB    = (const float*)d_in[10];
    const float* fW1 = (const float*)d_in[11]; const float* fb1 = (const float*)d_in[12];
    const float* fW2 = (const float*)d_in[13]; const float* fb2 = (const float*)d_in[14];
    const float* fW3 = (const float*)d_in[15]; const float* fb3 = (const float*)d_in[16];
    const float* hW1 = (const float*)d_in[17]; const float* hb1 = (const float*)d_in[18];
    const float* hW2 = (const float*)d_in[19]; const float* hb2 = (const float*)d_in[20];
    const float* hW3 = (const float*)d_in[21]; const float* hb3 = (const float*)d_in[22];
    const float* gW1 = (const float*)d_in[23]; const float* gb1 = (const float*)d_in[24];
    const float* gW2 = (const float*)d_in[25]; const float* gb2 = (const float*)d_in[26];
    const float* gW3 = (const float*)d_in[27]; const float* gb3 = (const float*)d_in[28];
    const float* pz0m = (const float*)d_in[29];
    const float* pz0l = (const float*)d_in[30];

    float* ws = (float*)d_ws;
    size_t o = 0;
    float* ctx  = ws + o; o += (size_t)T_STEPS * BATCH * CTXC;  // 134 MB
    float* h0   = ws + o; o += (size_t)BATCH * HID;
    float* h1   = ws + o; o += (size_t)BATCH * HID;
    float* gh   = ws + o; o += (size_t)BATCH * 3 * HID;
    float* qbuf = ws + o; o += (size_t)BATCH * 2 * LAT;
    float* zA   = ws + o; o += (size_t)BATCH * LAT;
    float* zB   = ws + o; o += (size_t)BATCH * LAT;
    float* cat  = ws + o; o += (size_t)BATCH * INF;
    float* m1   = ws + o; o += (size_t)BATCH * HID;
    float* m2   = ws + o; o += (size_t)BATCH * HID;
    float* fbuf = ws + o; o += (size_t)BATCH * LAT;
    float* hdrb = ws + o; o += (size_t)BATCH * LAT;
    float* gbuf = ws + o; o += (size_t)BATCH * LAT;
    float* alr  = ws + o; o += BATCH;
    float* apen = ws + o; o += BATCH;
    float* alp  = ws + o; o += BATCH;

    dim3 blk(128);
    // Dispatch: compile-time NT/activation; N%64==0 -> NT=4, else N==32 -> NT=2.
    auto gemm = [&](const float* A, int lda, const float* W, int ldw, const float* bias,
                    float* Cp, int ldc, int M, int N, int K, int act) {
        if ((N & 63) == 0) {
            dim3 grid(N / 64, M / 64);
            if (act) gemm_tile<4, true ><<<grid, blk, 0, stream>>>(A, lda, W, ldw, bias, Cp, ldc, K);
            else     gemm_tile<4, false><<<grid, blk, 0, stream>>>(A, lda, W, ldw, bias, Cp, ldc, K);
        } else {
            dim3 grid(1, M / 64);
            if (act) gemm_tile<2, true ><<<grid, blk, 0, stream>>>(A, lda, W, ldw, bias, Cp, ldc, K);
            else     gemm_tile<2, false><<<grid, blk, 0, stream>>>(A, lda, W, ldw, bias, Cp, ldc, K);
        }
    };

    // ---- GRU over reversed sequence; encoder fused into the loop ----
    zero_buf<<<(BATCH * HID + 255) / 256, 256, 0, stream>>>(h0, BATCH * HID);
    float* hc = h0; float* hn = h1;
    for (int t = 0; t < T_STEPS; ++t) {
        gemm(hc, HID, Whh, HID, bhh, gh, 3 * HID, BATCH, 3 * HID, HID, 0);
        gru_gate<<<(BATCH * HID + 255) / 256, 256, 0, stream>>>(xs, t, gh, hc, Wih, bih, hn);
        gemm(hn, HID, encW, HID, encB, ctx + (size_t)(T_STEPS - 1 - t) * BATCH * CTXC,
             CTXC, BATCH, CTXC, HID, 0);
        float* tmp = hc; hc = hn; hn = tmp;
    }

    // ---- q = ctx[0] @ qz0_W.T + b ; sample z0 ----
    gemm(ctx, CTXC, qW, CTXC, qB, qbuf, 2 * LAT, BATCH, 2 * LAT, CTXC, 0);
    init_z0<<<(BATCH * LAT + 255) / 256, 256, 0, stream>>>(qbuf, zA, alr, apen, alp, xs, noise);

    // ---- SDE scan ----
    float* zc = zA; float* zn = zB;
    for (int i = 0; i < T_STEPS - 1; ++i) {
        const float* ctx_t = ctx + (size_t)(i + 1) * BATCH * CTXC;
        concat_zc<<<(BATCH * INF + 255) / 256, 256, 0, stream>>>(zc, ctx_t, cat);
        // f = MLP3([z|ctx])
        gemm(cat, INF, fW1, INF, fb1, m1, HID, BATCH, HID, INF, 1);
        gemm(m1, HID, fW2, HID, fb2, m2, HID, BATCH, HID, HID, 1);
        gemm(m2, HID, fW3, HID, fb3, fbuf, LAT, BATCH, LAT, HID, 0);
        // hdr = MLP3(z)
        gemm(zc, LAT, hW1, LAT, hb1, m1, HID, BATCH, HID, LAT, 1);
        gemm(m1, HID, hW2, HID, hb2, m2, HID, BATCH, HID, HID, 1);
        gemm(m2, HID, hW3, HID, hb3, hdrb, LAT, BATCH, LAT, HID, 0);
        // g diag
        gdiag_kernel<<<(BATCH * LAT + 255) / 256, 256, 0, stream>>>(zc, gW1, gb1, gW2, gb2, gW3, gb3, gbuf);
        // Euler-Maruyama + accumulators
        sde_combine<<<(BATCH + 255) / 256, 256, 0, stream>>>(fbuf, hdrb, gbuf, zc, zn,
                                                             ts, i, xs, noise, alr, apen, alp);
        float* tmp = zc; zc = zn; zn = tmp;
    }

    final_reduce<<<1, 256, 0, stream>>>(alp, alr, apen, qbuf, pz0m, pz0l, ts, (float*)d_out);
}